// RatingLayer_84851373900497
// MI455X (gfx1250) — compile-verified
//
#include <hip/hip_runtime.h>

typedef __attribute__((ext_vector_type(2))) float v2f;
typedef __attribute__((ext_vector_type(8))) float v8f;

#define D 64
#define TILE 16

// ws (int32) layout:
//   [0  .. 15]  counts[r]
//   [16 .. 31]  offsets[r] (exclusive prefix of counts; offsets[R]=N)
//   [32 .. 47]  tile_offsets[r] (exclusive prefix of ceil(counts/16); [R]=total tiles)
//   [48 .. 63]  cursors[r]
//   [64 .. 64+N) perm: edge indices grouped by rating

__global__ void init_ws_kernel(int* ws) {
    int i = threadIdx.x;
    if (i < 64) ws[i] = 0;
}

__global__ void hist_kernel(const int* __restrict__ rating, int n, int* ws) {
    int e = blockIdx.x * blockDim.x + threadIdx.x;
    if (e < n) atomicAdd(&ws[rating[e]], 1);
}

__global__ void scan_kernel(int* ws, int R) {
    // single thread: tiny R-entry prefix sums
    int off = 0, toff = 0;
    for (int r = 0; r < R; ++r) {
        ws[16 + r] = off;
        ws[32 + r] = toff;
        ws[48 + r] = off;   // cursor starts at bucket base
        int c = ws[r];
        off  += c;
        toff += (c + TILE - 1) / TILE;
    }
    ws[16 + R] = off;
    ws[32 + R] = toff;
}

__global__ void scatter_kernel(const int* __restrict__ rating, int n, int* ws) {
    int e = blockIdx.x * blockDim.x + threadIdx.x;
    if (e < n) {
        int r = rating[e];
        int pos = atomicAdd(&ws[48 + r], 1);
        ws[64 + pos] = e;
    }
}

__global__ __launch_bounds__(256)
void rgcn_wmma_kernel(const float* __restrict__ embed,
                      const float* __restrict__ weights,
                      const float* __restrict__ inv_c,
                      const int*   __restrict__ edge_node,
                      float*       __restrict__ out,
                      const int*   __restrict__ ws, int R) {
    const int lane = threadIdx.x & 31;
    const int wave = threadIdx.x >> 5;
    const int tile = blockIdx.x * (blockDim.x >> 5) + wave;
    const int total_tiles = ws[32 + R];
    if (tile >= total_tiles) return;   // wave-uniform exit: EXEC stays all-1s for WMMA

    // map tile -> rating bucket (R <= 16, tiny scalar scan)
    int r = 0;
    while (r + 1 < R && ws[32 + r + 1] <= tile) ++r;
    const int tib          = tile - ws[32 + r];
    const int bucket_begin = ws[16 + r];
    const int bucket_end   = ws[16 + r + 1];
    const int base         = bucket_begin + tib * TILE;
    const int nvalid       = min(TILE, bucket_end - base);

    const int li  = lane & 15;            // row (A) / column (B,D) within 16-tile
    const int hi2 = (lane >> 4) << 1;     // K sub-offset: 0 for lanes 0-15, 2 for 16-31

    // per-lane edge for row li (clamped for partial tiles; masked at store)
    const int   e     = ws[64 + base + min(li, nvalid - 1)];
    const int   node  = edge_node[e];
    const float scale = inv_c[e];

    // A: lane holds embed[node][k + hi2 .. k + hi2 + 1]
    const float* arow  = embed + (size_t)node * D + hi2;
    // B[k][n] = W_r^T[k][n] = W_r[n][k]; lane holds W_r[li + 16t][k + hi2 .. +1]
    const float* wbase = weights + (size_t)r * (D * D) + (size_t)li * D + hi2;

    v8f a0 = {}, a1 = {}, a2 = {}, a3 = {};
#pragma unroll
    for (int k = 0; k < D; k += 4) {
        v2f a  = *(const v2f*)(arow + k);
        v2f b0 = *(const v2f*)(wbase + 0 * 16 * D + k);
        v2f b1 = *(const v2f*)(wbase + 1 * 16 * D + k);
        v2f b2 = *(const v2f*)(wbase + 2 * 16 * D + k);
        v2f b3 = *(const v2f*)(wbase + 3 * 16 * D + k);
        a0 = __builtin_amdgcn_wmma_f32_16x16x4_f32(false, a, false, b0, (short)0, a0, false, false);
        a1 = __builtin_amdgcn_wmma_f32_16x16x4_f32(false, a, false, b1, (short)0, a1, false, false);
        a2 = __builtin_amdgcn_wmma_f32_16x16x4_f32(false, a, false, b2, (short)0, a2, false, false);
        a3 = __builtin_amdgcn_wmma_f32_16x16x4_f32(false, a, false, b3, (short)0, a3, false, false);
    }

    // D tile layout: VGPR v holds (M = v, N = li) for lanes 0-15, (M = v+8, N = li) for 16-31
#pragma unroll
    for (int v = 0; v < 8; ++v) {
        const int src   = (lane < 16) ? v : (v + 8);     // tile row this lane stores
        const int onode = __shfl(node,  src, 32);
        const float osc = __shfl(scale, src, 32);
        if (src < nvalid) {
            float* orow = out + (size_t)onode * D + li;
            orow[0]  = a0[v] * osc;
            orow[16] = a1[v] * osc;
            orow[32] = a2[v] * osc;
            orow[48] = a3[v] * osc;
        }
    }
}

extern "C" void kernel_launch(void* const* d_in, const int* in_sizes, int n_in,
                              void* d_out, int out_size, void* d_ws, size_t ws_size,
                              hipStream_t stream) {
    const float* embed       = (const float*)d_in[0];
    const float* weights     = (const float*)d_in[1];
    const float* inv_c       = (const float*)d_in[2];
    const int*   edge_node   = (const int*)d_in[3];
    const int*   edge_rating = (const int*)d_in[4];
    float*       out         = (float*)d_out;
    int*         ws          = (int*)d_ws;

    const int n = in_sizes[2];               // number of edges (== nodes)
    const int R = in_sizes[1] / (D * D);     // number of ratings (10)

    init_ws_kernel<<<1, 64, 0, stream>>>(ws);
    hist_kernel<<<(n + 255) / 256, 256, 0, stream>>>(edge_rating, n, ws);
    scan_kernel<<<1, 1, 0, stream>>>(ws, R);
    scatter_kernel<<<(n + 255) / 256, 256, 0, stream>>>(edge_rating, n, ws);

    const int max_tiles       = (n + TILE - 1) / TILE + R;  // upper bound on total tiles
    const int waves_per_block = 8;                          // 256 threads = 8 wave32
    const int blocks          = (max_tiles + waves_per_block - 1) / waves_per_block;
    rgcn_wmma_kernel<<<blocks, 256, 0, stream>>>(embed, weights, inv_c, edge_node,
                                                 out, ws, R);
}